// ChiSquareLoss_17884243821445
// MI455X (gfx1250) — compile-verified
//
#include <hip/hip_runtime.h>

#define BINS 256
#define NCH 3
#define HW (512 * 512)                 // 262144 = 2^18
#define IMG_ELEMS (NCH * HW)           // 786432 elements per image
#define PB 64                          // blocks per image per tensor
#define BLK_ELEMS (IMG_ELEMS / PB)     // 12288 elements per block
#define THREADS 256
#define WAVES (THREADS / 32)           // 8 waves (wave32)

typedef float v4f __attribute__((ext_vector_type(4)));
typedef float v2f __attribute__((ext_vector_type(2)));
typedef float v8f __attribute__((ext_vector_type(8)));

// ---------------------------------------------------------------------------
// Kernel 1: per-image RGB histograms (u32 counts) for BOTH tensors.
// gridDim.x = nimg*PB (chunk of one image), gridDim.y = 2 (tensor select).
// Per-wave privatized LDS histograms -> ds_add_u32 with intra-wave-only
// contention; B128 non-temporal streaming loads (data read exactly once).
// ---------------------------------------------------------------------------
__global__ __launch_bounds__(THREADS) void chisq_hist_kernel(
    const float* __restrict__ x0, const float* __restrict__ x1,
    unsigned* __restrict__ hist, int nimg) {
  __shared__ unsigned lh[WAVES][NCH * BINS];   // 8 * 3KB = 24KB LDS
  const int tid  = threadIdx.x;
  const int wave = tid >> 5;                   // wave32

  for (int i = tid; i < WAVES * NCH * BINS; i += THREADS)
    (&lh[0][0])[i] = 0u;
  __syncthreads();

  const int tensor = blockIdx.y;
  const float* __restrict__ src = tensor ? x1 : x0;
  const int img   = blockIdx.x / PB;
  const int chunk = blockIdx.x % PB;
  const long long base = (long long)img * IMG_ELEMS + (long long)chunk * BLK_ELEMS;
  const v4f* __restrict__ p = (const v4f*)(src + base);
  const int lbase = chunk * BLK_ELEMS;         // element offset within image

#pragma unroll
  for (int k = 0; k < BLK_ELEMS / (THREADS * 4); ++k) {  // 12 iterations
    const int i4 = tid + k * THREADS;
    v4f f = __builtin_nontemporal_load(p + i4);          // B128, NT hint
    const int l    = lbase + i4 * 4;                     // HW % 4 == 0: no straddle
    const int coff = (l >> 18) << 8;                     // channel * 256
#pragma unroll
    for (int j = 0; j < 4; ++j) {
      int bi = (int)(f[j] * 255.0f);                     // trunc == floor (v >= 0)
      bi = bi < 0 ? 0 : (bi > 255 ? 255 : bi);           // matches jnp.clip
      atomicAdd(&lh[wave][coff + bi], 1u);               // ds_add_u32
    }
  }
  __syncthreads();

  unsigned* __restrict__ gh =
      hist + ((long long)tensor * nimg + img) * (NCH * BINS);
  for (int b = tid; b < NCH * BINS; b += THREADS) {
    unsigned s = 0;
#pragma unroll
    for (int w = 0; w < WAVES; ++w) s += lh[w][b];
    if (s) atomicAdd(&gh[b], s);                         // global_atomic_add_u32
  }
}

// ---------------------------------------------------------------------------
// Kernel 2: per-image chi-square over 768 bins. Normalization divisor is
// statically known: every element lands in a bin, so sum(hist) == IMG_ELEMS.
// ---------------------------------------------------------------------------
__global__ __launch_bounds__(THREADS) void chisq_reduce_kernel(
    const unsigned* __restrict__ hist, float* __restrict__ chi, int nimg) {
  __shared__ float sd[THREADS];
  const int b = blockIdx.x;
  const unsigned* __restrict__ h1 = hist + (long long)b * (NCH * BINS);
  const unsigned* __restrict__ h2 = hist + ((long long)nimg + b) * (NCH * BINS);
  const float invN = 1.0f / (float)IMG_ELEMS;

  float acc = 0.0f;
  for (int i = threadIdx.x; i < NCH * BINS; i += THREADS) {
    const float p1 = (float)h1[i] * invN;
    const float p2 = (float)h2[i] * invN;
    const float d  = p1 - p2;
    acc += d * d / (p1 + p2 + 1e-10f);
  }
  sd[threadIdx.x] = acc;
  __syncthreads();
  for (int s = THREADS / 2; s > 0; s >>= 1) {
    if (threadIdx.x < s) sd[threadIdx.x] += sd[threadIdx.x + s];
    __syncthreads();
  }
  if (threadIdx.x == 0) chi[b] = sd[0];
}

// ---------------------------------------------------------------------------
// Kernel 3: batch mean of the 32 per-image chi values via one
// V_WMMA_F32_16X16X4_F32. With A == all-ones, every row of D equals the
// column-sums of B, so summing one D row gives sum(all entries of B) —
// correct for ANY A/B VGPR layout. B carries {chi[lane], 0} per lane
// (32 values + 32 zeros over the 64 B slots). Single full wave, no
// divergence before the WMMA -> EXEC all-1s as required.
// chi[] is zero-initialized for 32 slots, so unconditional loads are safe.
// ---------------------------------------------------------------------------
__global__ __launch_bounds__(32) void chisq_mean_kernel(
    const float* __restrict__ chi, float* __restrict__ out, int nimg) {
  const int lane = threadIdx.x;
  const float v = chi[lane];

  v2f a; a[0] = 1.0f; a[1] = 1.0f;   // A: all ones (layout-independent)
  v2f b; b[0] = v;    b[1] = 0.0f;   // B: each chi value exactly once
  v8f c = {};
  c = __builtin_amdgcn_wmma_f32_16x16x4_f32(
      /*neg_a=*/false, a, /*neg_b=*/false, b,
      /*c_mod=*/(short)0, c, /*reuse_a=*/false, /*reuse_b=*/false);

  // Lanes 0-15 hold D[0][n], lanes 16-31 hold D[8][n]; both rows equal the
  // column sums, so a full 32-lane reduce yields 2 * sum(chi).
  float s = c[0];
#pragma unroll
  for (int off = 16; off >= 1; off >>= 1) s += __shfl_xor(s, off, 32);
  if (lane == 0) out[0] = s * (1.0f / (2.0f * (float)nimg));
}

// ---------------------------------------------------------------------------
extern "C" void kernel_launch(void* const* d_in, const int* in_sizes, int n_in,
                              void* d_out, int out_size, void* d_ws, size_t ws_size,
                              hipStream_t stream) {
  const float* x0 = (const float*)d_in[0];
  const float* x1 = (const float*)d_in[1];
  const int nimg = in_sizes[0] / IMG_ELEMS;   // 32

  unsigned* hist = (unsigned*)d_ws;
  const size_t histBytes = (size_t)2 * nimg * NCH * BINS * sizeof(unsigned);
  float* chi = (float*)((char*)d_ws + histBytes);

  // Zero counts + 32 chi slots each call (graph-capturable memset node).
  hipMemsetAsync(d_ws, 0, histBytes + 32 * sizeof(float), stream);

  dim3 grid(nimg * PB, 2);
  chisq_hist_kernel<<<grid, THREADS, 0, stream>>>(x0, x1, hist, nimg);
  chisq_reduce_kernel<<<nimg, THREADS, 0, stream>>>(hist, chi, nimg);
  chisq_mean_kernel<<<1, 32, 0, stream>>>(chi, (float*)d_out, nimg);
}